// LiteralE_42451456754033
// MI455X (gfx1250) — compile-verified
//
#include <hip/hip_runtime.h>
#include <math.h>
#include <stdint.h>

// LiteralE gated update: out = Z*Hc + (1-Z)*e_idx
//   Z  = sigmoid(W_ze[idx]@e + W_zl[idx]@l + b)
//   Hc = tanh   (W_he[idx]@e + W_hl[idx]@l)
// HBM-bound (~192KB gathered weights per batch element). Weights are streamed
// with async global->LDS b128 copies (double-buffered, ASYNCcnt-paced), and
// the mat-vecs run on V_WMMA_F32_16X16X4_F32 with a broadcast-B N=1 mapping.

#define EMB 128
#define FEA 64
#define SE  132   // padded LDS row stride for 128-col slabs (conflict-free b64)
#define SF  68    // padded LDS row stride for 64-col slabs

typedef __attribute__((ext_vector_type(2))) float v2f;
typedef __attribute__((ext_vector_type(8))) float v8f;

// Builtin signature (from hipcc diagnostic): param0 = v4i in AS(1) (global),
// param1 = LDS dst, then imm offset + cpol.
typedef int v4i_vs __attribute__((__vector_size__(16)));
typedef v4i_vs __attribute__((address_space(1)))* gptr_v4i;
typedef v4i_vs __attribute__((address_space(3)))* lptr_v4i;

struct alignas(16) Slab {
    float aze[16][SE];
    float ahe[16][SE];
    float azl[16][SF];
    float ahl[16][SF];
};

__device__ __forceinline__ void async_cp16(const float* g, float* s) {
    __builtin_amdgcn_global_load_async_to_lds_b128(
        (gptr_v4i)(uintptr_t)g,
        (lptr_v4i)(uint32_t)(uintptr_t)s,
        0, 0);
}

template <int N> __device__ __forceinline__ void wait_async() {
#if __has_builtin(__builtin_amdgcn_s_wait_asynccnt)
    __builtin_amdgcn_s_wait_asynccnt(N);
#else
    asm volatile("s_wait_asynccnt %0" :: "n"(N) : "memory");
#endif
}

// Each wave issues 12 async row-copies for row-block rb into slab sb.
__device__ __forceinline__ void stage(Slab* sb,
                                      const float* WzeB, const float* WheB,
                                      const float* WzlB, const float* WhlB,
                                      int rb, int wave, int lane) {
    // 512B rows (K=128): wave w stages rows 4w..4w+3 of W_ze and W_he.
    #pragma unroll
    for (int r4 = 0; r4 < 4; ++r4) {
        const int r   = wave * 4 + r4;
        const int row = rb * 16 + r;
        async_cp16(WzeB + (size_t)row * EMB + lane * 4, &sb->aze[r][lane * 4]);
        async_cp16(WheB + (size_t)row * EMB + lane * 4, &sb->ahe[r][lane * 4]);
    }
    // 256B rows (K=64): one b128 instr covers two rows (halves of the wave).
    #pragma unroll
    for (int p2 = 0; p2 < 2; ++p2) {
        const int r   = wave * 4 + p2 * 2 + (lane >> 4);
        const int c   = (lane & 15) * 4;
        const int row = rb * 16 + r;
        async_cp16(WzlB + (size_t)row * FEA + c, &sb->azl[r][c]);
        async_cp16(WhlB + (size_t)row * FEA + c, &sb->ahl[r][c]);
    }
}

__global__ __launch_bounds__(128) void literal_e_kernel(
    const float* __restrict__ e,   const float* __restrict__ l,
    const float* __restrict__ Wze, const float* __restrict__ Wzl,
    const float* __restrict__ Whe, const float* __restrict__ Whl,
    const float* __restrict__ b,   const long long* __restrict__ index,
    float* __restrict__ out)
{
    __shared__ Slab  slabs[2];
    __shared__ float xe[EMB];
    __shared__ float xl[FEA];
    __shared__ float bsh[EMB];
    __shared__ float zsh[EMB];
    __shared__ float hsh[EMB];

    const int t = threadIdx.x;
    const int n = blockIdx.x;
    const long long idx = index[n];

    const float* WzeB = Wze + (size_t)idx * EMB * EMB;
    const float* WheB = Whe + (size_t)idx * EMB * EMB;
    const float* WzlB = Wzl + (size_t)idx * EMB * FEA;
    const float* WhlB = Whl + (size_t)idx * EMB * FEA;

    const int lane = t & 31;
    const int wave = t >> 5;
    const int half = lane >> 4;    // selects K pair {0,1} vs {2,3} of a tile
    const int m    = lane & 15;    // A-tile row

    // Stage vectors/bias, zero shared accumulators, kick off slab 0.
    xe[t]  = e[(size_t)idx * EMB + t];
    bsh[t] = b[t];
    zsh[t] = 0.0f;
    hsh[t] = 0.0f;
    if (t < FEA) xl[t] = l[(size_t)idx * FEA + t];
    stage(&slabs[0], WzeB, WheB, WzlB, WhlB, 0, wave, lane);

    for (int rb = 0; rb < 8; ++rb) {
        Slab* cur = &slabs[rb & 1];
        if (rb < 7) {
            stage(&slabs[(rb + 1) & 1], WzeB, WheB, WzlB, WhlB, rb + 1, wave, lane);
            wait_async<12>();   // in-order completion => rb's 12 copies done
        } else {
            wait_async<0>();
        }
        __syncthreads();        // whole slab (all waves' copies) visible

        v8f accz = {0.f,0.f,0.f,0.f,0.f,0.f,0.f,0.f};
        v8f acch = {0.f,0.f,0.f,0.f,0.f,0.f,0.f,0.f};

        // K split 4 ways across waves: EMB slice of 32, FEA slice of 16.
        const int kw = wave * 32;
        #pragma unroll 4
        for (int k = 0; k < 32; k += 4) {
            const int k0 = kw + k + 2 * half;
            v2f a1 = *(const v2f*)&cur->aze[m][k0];
            v2f a2 = *(const v2f*)&cur->ahe[m][k0];
            v2f xb = *(const v2f*)&xe[k0];
            accz = __builtin_amdgcn_wmma_f32_16x16x4_f32(
                       false, a1, false, xb, (short)0, accz, false, false);
            acch = __builtin_amdgcn_wmma_f32_16x16x4_f32(
                       false, a2, false, xb, (short)0, acch, false, false);
        }
        const int kw2 = wave * 16;
        #pragma unroll 4
        for (int k = 0; k < 16; k += 4) {
            const int k0 = kw2 + k + 2 * half;
            v2f a1 = *(const v2f*)&cur->azl[m][k0];
            v2f a2 = *(const v2f*)&cur->ahl[m][k0];
            v2f xb = *(const v2f*)&xl[k0];
            accz = __builtin_amdgcn_wmma_f32_16x16x4_f32(
                       false, a1, false, xb, (short)0, accz, false, false);
            acch = __builtin_amdgcn_wmma_f32_16x16x4_f32(
                       false, a2, false, xb, (short)0, acch, false, false);
        }

        // Every D column identical (broadcast B): lanes 0/16 hold the 16 rows.
        if (m == 0) {
            const int rowbase = rb * 16 + half * 8;
            #pragma unroll
            for (int r = 0; r < 8; ++r) {
                atomicAdd(&zsh[rowbase + r], accz[r]);   // ds_add_f32
                atomicAdd(&hsh[rowbase + r], acch[r]);
            }
        }
        __syncthreads();        // slab consumed; safe to overwrite next round
    }

    const float zv = 1.0f / (1.0f + expf(-(zsh[t] + bsh[t])));
    const float hv = tanhf(hsh[t]);
    out[(size_t)n * EMB + t] = zv * hv + (1.0f - zv) * xe[t];
}

extern "C" void kernel_launch(void* const* d_in, const int* in_sizes, int n_in,
                              void* d_out, int out_size, void* d_ws, size_t ws_size,
                              hipStream_t stream) {
    const float*     e   = (const float*)d_in[0];
    const float*     l   = (const float*)d_in[1];
    const float*     Wze = (const float*)d_in[2];
    const float*     Wzl = (const float*)d_in[3];
    const float*     Whe = (const float*)d_in[4];
    const float*     Whl = (const float*)d_in[5];
    const float*     b   = (const float*)d_in[6];
    const long long* idx = (const long long*)d_in[7];
    float* out = (float*)d_out;

    const int batch = in_sizes[7];  // 8192
    literal_e_kernel<<<batch, 128, 0, stream>>>(e, l, Wze, Wzl, Whe, Whl, b, idx, out);
}